// GCNExplainer_1709396984302
// MI455X (gfx1250) — compile-verified
//
#include <hip/hip_runtime.h>

// ---------------------------------------------------------------------------
// 2-layer GCN (PyG GCNConv semantics) for MI455X / gfx1250.
//   F_IN=128, F_HID=64, F_OUT=16, N=50000 (divisible by 16), E=800000.
// GEMMs use native f32 WMMA (V_WMMA_F32_16X16X4_F32) -> exact f32 accumulate.
// Edge aggregation uses hardware global_atomic_add_f32 (unsafeAtomicAdd).
// ---------------------------------------------------------------------------

typedef float v2f __attribute__((ext_vector_type(2)));
typedef float v8f __attribute__((ext_vector_type(8)));

#define F_IN  128
#define F_HID 64
#define F_OUT 16

// ---- degree / normalization ------------------------------------------------
__global__ void k_init_deg(float* __restrict__ deg, int n) {
    int i = blockIdx.x * blockDim.x + threadIdx.x;
    if (i < n) deg[i] = 1.0f;                       // self-loop contribution
}

__global__ void k_accum_deg(const int* __restrict__ dst, float* __restrict__ deg, int E) {
    int i = blockIdx.x * blockDim.x + threadIdx.x;
    if (i < E) unsafeAtomicAdd(&deg[dst[i]], 1.0f); // global_atomic_add_f32
}

__global__ void k_rsqrt_inplace(float* __restrict__ deg, int n) {
    int i = blockIdx.x * blockDim.x + threadIdx.x;
    if (i < n) deg[i] = __frsqrt_rn(deg[i]);        // deg >= 1 always (self-loops)
}

// ---- GEMM1: h1[N,64] = x[N,128] @ W1[128,64], f32 WMMA ---------------------
// Block = 256 threads = 8 waves; each wave computes one 16x16 output tile.
// grid = (ceil(mTiles/8), 4 n-tiles). B tile staged in LDS, shared by 8 waves.
__global__ __launch_bounds__(256)
void k_gemm1(const float* __restrict__ X, const float* __restrict__ W,
             float* __restrict__ H, int mTiles) {
    __shared__ float Bs[F_IN * 16];                 // 8 KB
    const int nBase = blockIdx.y * 16;
    for (int idx = threadIdx.x; idx < F_IN * 16; idx += 256) {
        int k = idx >> 4, nn = idx & 15;
        Bs[idx] = W[k * F_HID + nBase + nn];
    }
    __syncthreads();

    const int wave  = threadIdx.x >> 5;
    const int lane  = threadIdx.x & 31;
    const int mTile = blockIdx.x * 8 + wave;
    if (mTile >= mTiles) return;
    const int mBase = mTile * 16;
    const int half  = lane >> 4;                    // 0: K pair {0,1}, 1: {2,3}
    const int l     = lane & 15;
    const float* arow = X + (size_t)(mBase + l) * F_IN;

    v8f c = {};
    #pragma unroll 4
    for (int k = 0; k < F_IN; k += 4) {
        const int ak = k + half * 2;
        v2f a = *(const v2f*)(arow + ak);           // global_load_b64
        v2f b; b.x = Bs[ak * 16 + l]; b.y = Bs[(ak + 1) * 16 + l];
        c = __builtin_amdgcn_wmma_f32_16x16x4_f32(false, a, false, b,
                                                  (short)0, c, false, false);
    }
    float* orow = H + (size_t)(mBase + half * 8) * F_HID + nBase + l;
    #pragma unroll
    for (int r = 0; r < 8; ++r) orow[(size_t)r * F_HID] = c[r];
}

// ---- layer-1 aggregation ---------------------------------------------------
__global__ void k_self_init64(const float* __restrict__ h1, const float* __restrict__ dinv,
                              float* __restrict__ agg, int n) {
    int t = blockIdx.x * blockDim.x + threadIdx.x;
    if (t < n * F_HID) {
        int i = t >> 6;                             // node
        float di = dinv[i];
        agg[t] = h1[t] * di * di;                   // self-loop: norm = dinv^2
    }
}

// One wave32 per edge: 64 features as float2 per lane, 2 f32 atomics per lane.
__global__ __launch_bounds__(256)
void k_scatter64(const int* __restrict__ src, const int* __restrict__ dst,
                 const float* __restrict__ h1, const float* __restrict__ dinv,
                 float* __restrict__ agg, int E) {
    int warp = (blockIdx.x * blockDim.x + threadIdx.x) >> 5;
    int lane = threadIdx.x & 31;
    if (warp >= E) return;
    int s = src[warp], d = dst[warp];
    float norm = dinv[s] * dinv[d];
    float2 v = ((const float2*)(h1 + (size_t)s * F_HID))[lane];
    float* drow = agg + (size_t)d * F_HID + 2 * lane;
    unsafeAtomicAdd(drow + 0, v.x * norm);
    unsafeAtomicAdd(drow + 1, v.y * norm);
}

// ---- GEMM2: h2[N,16] = relu(agg1 + b1)[N,64] @ W2[64,16], fused ReLU ------
__global__ __launch_bounds__(256)
void k_gemm2(const float* __restrict__ A1, const float* __restrict__ bias1,
             const float* __restrict__ W, float* __restrict__ H, int mTiles) {
    __shared__ float Bs[F_HID * 16];                // 4 KB
    __shared__ float b1s[F_HID];
    for (int idx = threadIdx.x; idx < F_HID * 16; idx += 256) {
        int k = idx >> 4, nn = idx & 15;
        Bs[idx] = W[k * F_OUT + nn];                // single 16-wide n tile
    }
    if (threadIdx.x < F_HID) b1s[threadIdx.x] = bias1[threadIdx.x];
    __syncthreads();

    const int wave  = threadIdx.x >> 5;
    const int lane  = threadIdx.x & 31;
    const int mTile = blockIdx.x * 8 + wave;
    if (mTile >= mTiles) return;
    const int mBase = mTile * 16;
    const int half  = lane >> 4;
    const int l     = lane & 15;
    const float* arow = A1 + (size_t)(mBase + l) * F_HID;

    v8f c = {};
    #pragma unroll 4
    for (int k = 0; k < F_HID; k += 4) {
        const int ak = k + half * 2;
        v2f a, b;
        a.x = fmaxf(arow[ak]     + b1s[ak],     0.0f);   // fused bias + ReLU
        a.y = fmaxf(arow[ak + 1] + b1s[ak + 1], 0.0f);
        b.x = Bs[ak * 16 + l]; b.y = Bs[(ak + 1) * 16 + l];
        c = __builtin_amdgcn_wmma_f32_16x16x4_f32(false, a, false, b,
                                                  (short)0, c, false, false);
    }
    float* orow = H + (size_t)(mBase + half * 8) * F_OUT + l;
    #pragma unroll
    for (int r = 0; r < 8; ++r) orow[(size_t)r * F_OUT] = c[r];
}

// ---- layer-2 aggregation (writes d_out directly) ---------------------------
__global__ void k_self_init16(const float* __restrict__ h2, const float* __restrict__ dinv,
                              const float* __restrict__ bias2, float* __restrict__ out, int n) {
    int t = blockIdx.x * blockDim.x + threadIdx.x;
    if (t < n * F_OUT) {
        int i = t >> 4, f = t & 15;
        float di = dinv[i];
        out[t] = h2[t] * di * di + bias2[f];
    }
}

// 16 lanes per edge, one feature per lane.
__global__ __launch_bounds__(256)
void k_scatter16(const int* __restrict__ src, const int* __restrict__ dst,
                 const float* __restrict__ h2, const float* __restrict__ dinv,
                 float* __restrict__ out, int E) {
    int t = blockIdx.x * blockDim.x + threadIdx.x;
    int e = t >> 4, f = t & 15;
    if (e >= E) return;
    int s = src[e], d = dst[e];
    float norm = dinv[s] * dinv[d];
    unsafeAtomicAdd(&out[(size_t)d * F_OUT + f], h2[(size_t)s * F_OUT + f] * norm);
}

// ---------------------------------------------------------------------------
extern "C" void kernel_launch(void* const* d_in, const int* in_sizes, int n_in,
                              void* d_out, int out_size, void* d_ws, size_t ws_size,
                              hipStream_t stream) {
    const float* x  = (const float*)d_in[0];
    const int*   ei = (const int*)  d_in[1];
    const float* W1 = (const float*)d_in[2];
    const float* b1 = (const float*)d_in[3];
    const float* W2 = (const float*)d_in[4];
    const float* b2 = (const float*)d_in[5];
    float* out = (float*)d_out;

    const int n = in_sizes[0] / F_IN;       // 50000
    const int E = in_sizes[1] / 2;          // 800000
    const int* src = ei;
    const int* dst = ei + E;

    // workspace layout (f32): dinv[n] | h1[n*64] | agg1[n*64] | h2[n*16]
    float* dinv = (float*)d_ws;
    float* h1   = dinv + n;
    float* agg1 = h1 + (size_t)n * F_HID;
    float* h2   = agg1 + (size_t)n * F_HID;

    const int mTiles  = n / 16;             // 3125 (exact)
    const int mGroups = (mTiles + 7) / 8;

    // 1) symmetric normalization coefficients
    k_init_deg<<<(n + 255) / 256, 256, 0, stream>>>(dinv, n);
    k_accum_deg<<<(E + 255) / 256, 256, 0, stream>>>(dst, dinv, E);
    k_rsqrt_inplace<<<(n + 255) / 256, 256, 0, stream>>>(dinv, n);

    // 2) h1 = x @ W1   (WMMA f32)
    k_gemm1<<<dim3(mGroups, F_HID / 16), 256, 0, stream>>>(x, W1, h1, mTiles);

    // 3) agg1 = scatter(norm * h1) incl. self-loops
    k_self_init64<<<((size_t)n * F_HID + 255) / 256, 256, 0, stream>>>(h1, dinv, agg1, n);
    k_scatter64<<<(int)(((size_t)E * 32 + 255) / 256), 256, 0, stream>>>(src, dst, h1, dinv, agg1, E);

    // 4) h2 = relu(agg1 + b1) @ W2   (WMMA f32, fused bias+ReLU)
    k_gemm2<<<dim3(mGroups, 1), 256, 0, stream>>>(agg1, b1, W2, h2, mTiles);

    // 5) out = scatter(norm * h2) + b2 incl. self-loops
    k_self_init16<<<((size_t)n * F_OUT + 255) / 256, 256, 0, stream>>>(h2, dinv, b2, out, n);
    k_scatter16<<<(int)(((size_t)E * 16 + 255) / 256), 256, 0, stream>>>(src, dst, h2, dinv, out, E);
}